// Model_12601434046515
// MI455X (gfx1250) — compile-verified
//
#include <hip/hip_runtime.h>
#include <math.h>

// ---------------------------------------------------------------------------
// MI455X (gfx1250) implementation.
//
// Identity: the reference gathers node features at `dst` and scatter-means to
// the SAME `dst`, and every CG tensor-product term is linear in sh1(e):
//     scatter_mean_e TP(feat[dst], sh1_e)  ==  TP(feat[n], mean_e sh1_e)
// (bias suppressed for zero-in-degree nodes, matching the cnt clamp).
// => ONE edge pass (per-node mean sh + degree), then a node-wise pipeline.
//
// Edge pass:  ~25.6 MB streamed indices; positions (1.2 MB) and the 1.6 MB
//             accumulator are L2-resident -> a few microseconds at 23.3 TB/s.
// Node pass:  16 nodes per wave32; every linear layer runs on
//             V_WMMA_F32_16X16X4_F32 (exact f32 math).  All B fragments and
//             padded biases are pre-swizzled into LDS once per block, so the
//             inner loop is ds_load_b64 x2 + v_wmma, with no divergent loads.
// ---------------------------------------------------------------------------

typedef __attribute__((ext_vector_type(2))) float v2f;
typedef __attribute__((ext_vector_type(8))) float v8f;

#define SQRT3_F     1.7320508075688772f
#define INV_SQRT3_F 0.5773502691896258f
#define WPB 8  // waves (16-node tiles) per 256-thread block

__device__ __forceinline__ float gelu_exact(float x) {
  // torch GELU (exact erf form)
  return 0.5f * x * (1.0f + erff(x * 0.70710678118654752440f));
}

// ---------------- edge pass: per-node mean of sh1 + degree ------------------
__global__ void edge_mean_sh_kernel(const float* __restrict__ pos,
                                    const int* __restrict__ esrc,
                                    const int* __restrict__ edst,
                                    float* __restrict__ Msum,
                                    float* __restrict__ cnt,
                                    int E) {
  const int stride = blockDim.x * gridDim.x;
  for (int e = blockIdx.x * blockDim.x + threadIdx.x; e < E; e += stride) {
    const int ahead = e + 8 * stride;
    if (ahead < E) {  // global_prefetch_b8 on the streamed index arrays
      __builtin_prefetch(esrc + ahead, 0, 0);
      __builtin_prefetch(edst + ahead, 0, 0);
    }
    const int s = esrc[e];
    const int d = edst[e];
    const float dx = pos[3 * d + 0] - pos[3 * s + 0];
    const float dy = pos[3 * d + 1] - pos[3 * s + 1];
    const float dz = pos[3 * d + 2] - pos[3 * s + 2];
    const float r = sqrtf(dx * dx + dy * dy + dz * dz);
    const float w = SQRT3_F / fmaxf(r, 1e-9f);
    atomicAdd(&Msum[3 * d + 0], dx * w);
    atomicAdd(&Msum[3 * d + 1], dy * w);
    atomicAdd(&Msum[3 * d + 2], dz * w);
    atomicAdd(&cnt[d], 1.0f);
  }
}

// ---------------- WMMA helper: [16 x K] @ [K x O] in f32 --------------------
// A frag (16x4 f32): lane m=L&15, half=L>>4; VGPR0 holds K=k0+2*half,
// VGPR1 holds K=k0+2*half+1.  B frag (4x16): VGPR r holds K row, lane = N col,
// lane-half adds +2 to K (mirrors documented C/D row striping).
// D (16x16 f32): VGPR r -> node (r + 8*half), lane&15 -> output column.
// B fragments are pre-swizzled in LDS: bf[chunk*64 + lane*2 + {0,1}].
__device__ __forceinline__ v8f wmma_tile(const float* __restrict__ XA,  // LDS [16][12], zero padded
                                         const float* __restrict__ bf,  // LDS fragment table
                                         int nchunks, int lane) {
  const int mm = lane & 15;
  const int half = lane >> 4;
  v8f c = {};
  for (int cc = 0; cc < nchunks; ++cc) {
    const int kA = cc * 4 + 2 * half;
    v2f a, b;
    a.x = XA[mm * 12 + kA];          // ds_load_b64 (8B aligned: kA even)
    a.y = XA[mm * 12 + kA + 1];
    b.x = bf[cc * 64 + lane * 2 + 0];  // ds_load_b64
    b.y = bf[cc * 64 + lane * 2 + 1];
    c = __builtin_amdgcn_wmma_f32_16x16x4_f32(false, a, false, b, (short)0, c,
                                              false, false);
  }
  return c;
}

struct WaveScratch {
  float XA[16 * 12];  // GEMM A staging (zero padded to 12 cols)
  float m[16 * 3];    // mean sh per node
  float deg[16];      // 1 if in-degree > 0 else 0
  float sA[16 * 5];   // scalar feats (layer1 out -> gated; reused for s3)
  float vA[16 * 15];  // vector feats (layer1 out -> gated)
  float sB[16 * 5];   // layer2 scalar out -> gated
  float vB[16 * 15];  // layer2 vector out -> gated
};

// B fragment chunk table (21 chunks of 64 floats):
//  0..1  Wg1   (sO=5,  sK=1, O=10, K=5)
//  2..4  W0_2  (sO=10, sK=1, O=5,  K=10)
//  5..13 W1_2  component c=(ch-5)/3 (base W1_2+c, sO=30, sK=3, O=5, K=10)
// 14..15 Wg2   (sO=5,  sK=1, O=10, K=5)
// 16..18 W0_3  (sO=10, sK=1, O=5,  K=10)
// 19..20 Wg3   (sO=5,  sK=1, O=10, K=5)
#define NCHUNKS_TOTAL 21

// ---------------- node pipeline: 3 layers + gates, all WMMA -----------------
__global__ void __launch_bounds__(WPB * 32) node_pipeline_kernel(
    const float* __restrict__ Msum, const float* __restrict__ cnt,
    const float* __restrict__ W0_1, const float* __restrict__ W1_1,
    const float* __restrict__ b_1, const float* __restrict__ Wg1,
    const float* __restrict__ bg1, const float* __restrict__ W0_2,
    const float* __restrict__ W1_2, const float* __restrict__ b_2,
    const float* __restrict__ Wg2, const float* __restrict__ bg2,
    const float* __restrict__ W0_3, const float* __restrict__ b_3,
    const float* __restrict__ Wg3, const float* __restrict__ bg3,
    float* __restrict__ pooled, int N, int NT) {
  __shared__ WaveScratch sc[WPB];
  __shared__ float BF[NCHUNKS_TOTAL * 64];  // pre-swizzled B fragments
  __shared__ float BIAS[5 * 16];            // bg1,b_2,bg2,b_3,bg3 zero-padded
  __shared__ float L1s[5];                  // W0_1[o] + b_1[o]
  __shared__ float L1v[15];                 // W1_1
  __shared__ float pool_lds[5];

  const int lane = threadIdx.x & 31;
  const int wid = threadIdx.x >> 5;
  WaveScratch& S = sc[wid];
  const int mm = lane & 15;
  const int half = lane >> 4;

  int tile = blockIdx.x * WPB + wid;
  const bool active = (tile < NT);
  if (!active) tile = 0;  // clamp: redundant compute keeps barriers/EXEC uniform
  const int base = tile * 16;

  // ---- block init: swizzle weights into LDS fragment tables ----
  for (int ch = wid; ch < NCHUNKS_TOTAL; ch += WPB) {
    const float* W;
    int sO, sK, O, K, k0;
    if (ch < 2) {
      W = Wg1; sO = 5; sK = 1; O = 10; K = 5; k0 = ch * 4;
    } else if (ch < 5) {
      W = W0_2; sO = 10; sK = 1; O = 5; K = 10; k0 = (ch - 2) * 4;
    } else if (ch < 14) {
      const int c = (ch - 5) / 3;
      W = W1_2 + c; sO = 30; sK = 3; O = 5; K = 10; k0 = ((ch - 5) % 3) * 4;
    } else if (ch < 16) {
      W = Wg2; sO = 5; sK = 1; O = 10; K = 5; k0 = (ch - 14) * 4;
    } else if (ch < 19) {
      W = W0_3; sO = 10; sK = 1; O = 5; K = 10; k0 = (ch - 16) * 4;
    } else {
      W = Wg3; sO = 5; sK = 1; O = 10; K = 5; k0 = (ch - 19) * 4;
    }
    const int nn = lane & 15;
    const int kA = k0 + 2 * (lane >> 4);
    BF[ch * 64 + lane * 2 + 0] =
        (nn < O && kA < K) ? W[nn * sO + kA * sK] : 0.0f;
    BF[ch * 64 + lane * 2 + 1] =
        (nn < O && kA + 1 < K) ? W[nn * sO + (kA + 1) * sK] : 0.0f;
  }
  if (threadIdx.x < 80) {
    const int which = threadIdx.x >> 4;
    const int o = threadIdx.x & 15;
    const float* bp = (which == 0) ? bg1
                    : (which == 1) ? b_2
                    : (which == 2) ? bg2
                    : (which == 3) ? b_3 : bg3;
    const int O = (which == 1 || which == 3) ? 5 : 10;
    BIAS[threadIdx.x] = (o < O) ? bp[o] : 0.0f;
  }
  if (threadIdx.x < 5) {
    L1s[threadIdx.x] = W0_1[threadIdx.x] + b_1[threadIdx.x];
    pool_lds[threadIdx.x] = 0.0f;
  }
  if (threadIdx.x < 15) L1v[threadIdx.x] = W1_1[threadIdx.x];
  __syncthreads();

  // ---- phase 1: mean sh + degree; layer1 closed form (input scalar == 1)
  if (lane < 16) {
    int node = base + mm;
    const bool ok = (node < N);
    if (!ok) node = 0;
    const float c = ok ? cnt[node] : 0.0f;
    const float dmask = (c > 0.0f) ? 1.0f : 0.0f;
    const float ic = 1.0f / fmaxf(c, 1.0f);
    float mv[3];
    for (int k = 0; k < 3; ++k) {
      mv[k] = ok ? Msum[3 * node + k] * ic : 0.0f;
      S.m[mm * 3 + k] = mv[k];
    }
    S.deg[mm] = dmask;
    for (int o = 0; o < 5; ++o) {
      S.sA[mm * 5 + o] = dmask * L1s[o];
      for (int k = 0; k < 3; ++k)
        S.vA[mm * 15 + o * 3 + k] = dmask * L1v[o * 3 + k] * mv[k];
    }
    for (int j = 0; j < 12; ++j)
      S.XA[mm * 12 + j] = (j < 5) ? S.sA[mm * 5 + j] : 0.0f;
  }
  __syncthreads();

  // ---- gate1: G1 = gelu(S1 @ Wg1^T + bg1); gate scalars/vectors in place
  {
    v8f g = wmma_tile(S.XA, &BF[0 * 64], 2, lane);
    const float bb = BIAS[0 * 16 + mm];
    float gv[8];
    for (int r = 0; r < 8; ++r) gv[r] = gelu_exact(g[r] + bb);
    __syncthreads();
    if (mm < 5) {
      for (int r = 0; r < 8; ++r) S.sA[(r + 8 * half) * 5 + mm] *= gv[r];
    } else if (mm < 10) {
      const int i = mm - 5;
      for (int r = 0; r < 8; ++r)
        for (int k = 0; k < 3; ++k)
          S.vA[(r + 8 * half) * 15 + i * 3 + k] *= gv[r];
    }
  }
  __syncthreads();

  // ---- layer2 scalars: X2 = [s1g, (v1g.m)/sqrt3]; S2 = deg*(X2@W0_2^T + b2)
  if (lane < 16) {
    for (int i = 0; i < 5; ++i) S.XA[mm * 12 + i] = S.sA[mm * 5 + i];
    for (int i = 0; i < 5; ++i) {
      float dot = 0.0f;
      for (int k = 0; k < 3; ++k)
        dot += S.vA[mm * 15 + i * 3 + k] * S.m[mm * 3 + k];
      S.XA[mm * 12 + 5 + i] = dot * INV_SQRT3_F;
    }
    S.XA[mm * 12 + 10] = 0.0f;
    S.XA[mm * 12 + 11] = 0.0f;
  }
  __syncthreads();
  {
    v8f s2 = wmma_tile(S.XA, &BF[2 * 64], 3, lane);
    const float bb = BIAS[1 * 16 + mm];
    if (mm < 5)
      for (int r = 0; r < 8; ++r) {
        const int nl = r + 8 * half;
        S.sB[nl * 5 + mm] = S.deg[nl] * (s2[r] + bb);
      }
  }
  __syncthreads();

  // ---- layer2 vectors: TPV_k = [s1g*m_k, v1g_k]; V2_k = deg*(TPV_k@W1_2k^T)
  for (int k = 0; k < 3; ++k) {
    if (lane < 16) {
      for (int i = 0; i < 5; ++i)
        S.XA[mm * 12 + i] = S.sA[mm * 5 + i] * S.m[mm * 3 + k];
      for (int i = 0; i < 5; ++i)
        S.XA[mm * 12 + 5 + i] = S.vA[mm * 15 + i * 3 + k];
      // cols 10,11 stay zero
    }
    __syncthreads();
    v8f vv = wmma_tile(S.XA, &BF[(5 + 3 * k) * 64], 3, lane);
    if (mm < 5)
      for (int r = 0; r < 8; ++r) {
        const int nl = r + 8 * half;
        S.vB[nl * 15 + mm * 3 + k] = S.deg[nl] * vv[r];
      }
    __syncthreads();
  }

  // ---- gate2
  if (lane < 16)
    for (int j = 0; j < 12; ++j)
      S.XA[mm * 12 + j] = (j < 5) ? S.sB[mm * 5 + j] : 0.0f;
  __syncthreads();
  {
    v8f g = wmma_tile(S.XA, &BF[14 * 64], 2, lane);
    const float bb = BIAS[2 * 16 + mm];
    float gv[8];
    for (int r = 0; r < 8; ++r) gv[r] = gelu_exact(g[r] + bb);
    __syncthreads();
    if (mm < 5) {
      for (int r = 0; r < 8; ++r) S.sB[(r + 8 * half) * 5 + mm] *= gv[r];
    } else if (mm < 10) {
      const int i = mm - 5;
      for (int r = 0; r < 8; ++r)
        for (int k = 0; k < 3; ++k)
          S.vB[(r + 8 * half) * 15 + i * 3 + k] *= gv[r];
    }
  }
  __syncthreads();

  // ---- layer3 scalars: X3 = [s2g, (v2g.m)/sqrt3]; S3 = deg*(X3@W0_3^T + b3)
  if (lane < 16) {
    for (int i = 0; i < 5; ++i) S.XA[mm * 12 + i] = S.sB[mm * 5 + i];
    for (int i = 0; i < 5; ++i) {
      float dot = 0.0f;
      for (int k = 0; k < 3; ++k)
        dot += S.vB[mm * 15 + i * 3 + k] * S.m[mm * 3 + k];
      S.XA[mm * 12 + 5 + i] = dot * INV_SQRT3_F;
    }
    S.XA[mm * 12 + 10] = 0.0f;
    S.XA[mm * 12 + 11] = 0.0f;
  }
  __syncthreads();
  {
    v8f s3 = wmma_tile(S.XA, &BF[16 * 64], 3, lane);
    const float bb = BIAS[3 * 16 + mm];
    if (mm < 5)
      for (int r = 0; r < 8; ++r) {
        const int nl = r + 8 * half;
        S.sA[nl * 5 + mm] = S.deg[nl] * (s3[r] + bb);  // reuse sA as s3
      }
  }
  __syncthreads();

  // ---- gate3 + pooling (sum of gelu(g3)*s3 over the tile's 16 nodes)
  if (lane < 16)
    for (int j = 0; j < 12; ++j)
      S.XA[mm * 12 + j] = (j < 5) ? S.sA[mm * 5 + j] : 0.0f;
  __syncthreads();
  {
    v8f g = wmma_tile(S.XA, &BF[19 * 64], 2, lane);
    const float bb = BIAS[4 * 16 + mm];
    if (active && mm < 5) {
      float part = 0.0f;
      for (int r = 0; r < 8; ++r)
        part += gelu_exact(g[r] + bb) * S.sA[(r + 8 * half) * 5 + mm];
      atomicAdd(&pool_lds[mm], part);
    }
  }
  __syncthreads();
  if (threadIdx.x < 5) atomicAdd(&pooled[threadIdx.x], pool_lds[threadIdx.x]);
}

// ---------------- finalize: mean over nodes, output linear, softmax ---------
__global__ void finalize_kernel(const float* __restrict__ pooled,
                                const float* __restrict__ Wout,
                                const float* __restrict__ bout,
                                float* __restrict__ out, float invN) {
  if (threadIdx.x == 0) {
    float p[5];
    for (int i = 0; i < 5; ++i) p[i] = pooled[i] * invN;
    float lg[10];
    float mx = -1e30f;
    for (int k = 0; k < 10; ++k) {
      float a = bout[k];
      for (int i = 0; i < 5; ++i) a += Wout[k * 5 + i] * p[i];
      lg[k] = a;
      mx = fmaxf(mx, a);
    }
    float ssum = 0.0f;
    for (int k = 0; k < 10; ++k) {
      lg[k] = expf(lg[k] - mx);
      ssum += lg[k];
    }
    const float inv = 1.0f / ssum;
    for (int k = 0; k < 10; ++k) out[k] = lg[k] * inv;
  }
}

extern "C" void kernel_launch(void* const* d_in, const int* in_sizes, int n_in,
                              void* d_out, int out_size, void* d_ws,
                              size_t ws_size, hipStream_t stream) {
  const float* pos  = (const float*)d_in[0];
  const int* esrc   = (const int*)d_in[1];
  const int* edst   = (const int*)d_in[2];
  const float* W0_1 = (const float*)d_in[3];
  const float* W1_1 = (const float*)d_in[4];
  const float* b_1  = (const float*)d_in[5];
  const float* Wg1  = (const float*)d_in[6];
  const float* bg1  = (const float*)d_in[7];
  const float* W0_2 = (const float*)d_in[8];
  const float* W1_2 = (const float*)d_in[9];
  const float* b_2  = (const float*)d_in[10];
  const float* Wg2  = (const float*)d_in[11];
  const float* bg2  = (const float*)d_in[12];
  const float* W0_3 = (const float*)d_in[13];
  const float* b_3  = (const float*)d_in[14];
  const float* Wg3  = (const float*)d_in[15];
  const float* bg3  = (const float*)d_in[16];
  const float* Wout = (const float*)d_in[17];
  const float* bout = (const float*)d_in[18];

  const int N = in_sizes[0] / 3;
  const int E = in_sizes[1];

  float* Msum   = (float*)d_ws;                 // [N][3]
  float* cnt    = Msum + (size_t)3 * N;         // [N]
  float* pooled = cnt + N;                      // [5]

  hipMemsetAsync(d_ws, 0, ((size_t)4 * N + 5) * sizeof(float), stream);

  int eblocks = (E + 255) / 256;
  if (eblocks > 2048) eblocks = 2048;
  edge_mean_sh_kernel<<<eblocks, 256, 0, stream>>>(pos, esrc, edst, Msum, cnt,
                                                   E);

  const int NT = (N + 15) / 16;
  const int nblocks = (NT + WPB - 1) / WPB;
  node_pipeline_kernel<<<nblocks, WPB * 32, 0, stream>>>(
      Msum, cnt, W0_1, W1_1, b_1, Wg1, bg1, W0_2, W1_2, b_2, Wg2, bg2, W0_3,
      b_3, Wg3, bg3, pooled, N, NT);

  finalize_kernel<<<1, 32, 0, stream>>>(pooled, Wout, bout, (float*)d_out,
                                        1.0f / (float)N);
}